// EctLayer_3917010174516
// MI455X (gfx1250) — compile-verified
//
#include <hip/hip_runtime.h>

typedef __attribute__((ext_vector_type(2))) float v2f;
typedef __attribute__((ext_vector_type(8))) float v8f;

#define SBITS 32                 // S thresholds
#define TDIR  32                 // T directions
#define BGR   64                 // B graphs
#define KSCALE (50.0f * 1.44269504088896340736f)   // 50 * log2(e)
#define EDGE_CHUNKS 48

// ---------------------------------------------------------------------------
// Zero the output accumulator (harness poisons d_out with 0xAA).
// ---------------------------------------------------------------------------
__global__ void ect_zero(float* __restrict__ out, int n) {
  int i = blockIdx.x * blockDim.x + threadIdx.x;
  if (i < n) out[i] = 0.0f;
}

// ---------------------------------------------------------------------------
// nh = x @ v via V_WMMA_F32_16X16X4_F32.
// One wave -> 16 rows x 32 cols of nh using two WMMAs (cols 0..15, 16..31).
// A layout (16x4 f32): lanes 0-15 hold K=0,1 in v[0],v[1]; lanes 16-31 hold
// K=2,3.  F=3, so K=3 is zero-padded.  B (4x16) mirrors that split.
// D layout: VGPR r -> (M=r, N=lane) for lanes 0-15, (M=8+r, N=lane-16) else.
// Full tiles (the common case: N % 16 == 0) take a wave-uniform fast path
// with unconditional stores; only a tail tile pays per-row predication.
// ---------------------------------------------------------------------------
__global__ void ect_heights_wmma(const float* __restrict__ x,
                                 const float* __restrict__ v,
                                 float* __restrict__ nh, int N) {
  int wave = blockIdx.x * (blockDim.x >> 5) + (threadIdx.x >> 5);
  int lane = threadIdx.x & 31;
  int row0 = wave * 16;
  if (row0 >= N) return;                 // uniform per wave: EXEC stays full

  int  m    = lane & 15;
  bool hi   = lane >= 16;
  bool full = (row0 + 16 <= N);          // wave-uniform
  int  row  = row0 + m;

  v2f a; a.x = 0.0f; a.y = 0.0f;
  if (row < N) {
    if (!hi) { a.x = x[row * 3 + 0]; a.y = x[row * 3 + 1]; }
    else     { a.x = x[row * 3 + 2]; /* K=3 padded 0 */ }
  }

  v2f b0, b1;
  if (!hi) {
    b0.x = v[0 * TDIR + m];       b0.y = v[1 * TDIR + m];
    b1.x = v[0 * TDIR + 16 + m];  b1.y = v[1 * TDIR + 16 + m];
  } else {
    b0.x = v[2 * TDIR + m];       b0.y = 0.0f;
    b1.x = v[2 * TDIR + 16 + m];  b1.y = 0.0f;
  }

  v8f c0 = {}; v8f c1 = {};
  c0 = __builtin_amdgcn_wmma_f32_16x16x4_f32(false, a, false, b0, (short)0, c0,
                                             false, false);
  c1 = __builtin_amdgcn_wmma_f32_16x16x4_f32(false, a, false, b1, (short)0, c1,
                                             false, false);

  int rbase = row0 + (hi ? 8 : 0);
  if (full) {
    // one scalar branch, 16 unconditional coalesced stores
#pragma unroll
    for (int r = 0; r < 8; ++r) {
      nh[(rbase + r) * TDIR + m]      = c0[r];
      nh[(rbase + r) * TDIR + 16 + m] = c1[r];
    }
  } else {
#pragma unroll
    for (int r = 0; r < 8; ++r) {
      int rr = rbase + r;
      if (rr < N) {
        nh[rr * TDIR + m]      = c0[r];
        nh[rr * TDIR + 16 + m] = c1[r];
      }
    }
  }
}

// ---------------------------------------------------------------------------
// Node ECT: lane = direction t, register accumulator over S thresholds.
// batch is sorted -> contiguous chunks; flush with global f32 atomics only on
// graph change (rare).
// ---------------------------------------------------------------------------
__global__ void ect_nodes(const float* __restrict__ nh,
                          const int* __restrict__ batch,
                          const float* __restrict__ lin,
                          float* __restrict__ out, int N) {
  int lane = threadIdx.x & 31;
  int wave = blockIdx.x * (blockDim.x >> 5) + (threadIdx.x >> 5);
  int numWaves = gridDim.x * (blockDim.x >> 5);
  int chunk = (N + numWaves - 1) / numWaves;
  int start = wave * chunk;
  int end   = min(start + chunk, N);
  if (start >= end) return;

  float d[SBITS];
#pragma unroll
  for (int s = 0; s < SBITS; ++s) d[s] = KSCALE * lin[s];

  float acc[SBITS];
#pragma unroll
  for (int s = 0; s < SBITS; ++s) acc[s] = 0.0f;

  int g = batch[start] & (BGR - 1);
  for (int n = start; n < end; ++n) {
    int gn = batch[n] & (BGR - 1);
    if (gn != g) {                       // uniform across wave (same n)
#pragma unroll
      for (int s = 0; s < SBITS; ++s) {
        atomicAdd(&out[(g * SBITS + s) * TDIR + lane], acc[s]);
        acc[s] = 0.0f;
      }
      g = gn;
    }
    float c = KSCALE * nh[n * TDIR + lane];
#pragma unroll
    for (int s = 0; s < SBITS; ++s) {
      // sigmoid(50*(lin_s - h)) = 1 / (1 + 2^(K*h - K*lin_s))
      acc[s] += __builtin_amdgcn_rcpf(1.0f + __builtin_amdgcn_exp2f(c - d[s]));
    }
  }
#pragma unroll
  for (int s = 0; s < SBITS; ++s)
    atomicAdd(&out[(g * SBITS + s) * TDIR + lane], acc[s]);
}

// ---------------------------------------------------------------------------
// Edge ECT (subtracted): edges are unsorted by graph, so each wave owns one
// direction t and a contiguous edge chunk; lane = threshold s.  Accumulate
// into a wave-private LDS slab [64 graphs][32 s] (8KB/wave, 64KB/block, no
// atomics needed), then flush negated partials to global with atomics.
// Unroll 2: two independent edge gather chains in flight per iteration.
// ---------------------------------------------------------------------------
__global__ void ect_edges(const float* __restrict__ nh,
                          const int* __restrict__ edge_index,
                          const int* __restrict__ batch,
                          const float* __restrict__ lin,
                          float* __restrict__ out, int N, int E) {
  __shared__ float lacc[8][BGR * SBITS];          // 64 KB, wave-private slices
  int lane = threadIdx.x & 31;                    // lane = s
  int wv   = threadIdx.x >> 5;
  int wave = blockIdx.x * (blockDim.x >> 5) + wv;
  int t        = wave & (TDIR - 1);               // this wave's direction
  int chunkIdx = wave >> 5;                       // TDIR == 32

  float* myacc = lacc[wv];
  for (int i = lane; i < BGR * SBITS; i += 32) myacc[i] = 0.0f;

  float dl = KSCALE * lin[lane];

  int chunkSize = (E + EDGE_CHUNKS - 1) / EDGE_CHUNKS;
  int e0 = chunkIdx * chunkSize;
  int e1 = min(e0 + chunkSize, E);
#pragma unroll 2
  for (int e = e0; e < e1; ++e) {
    int src = edge_index[e];
    int dst = edge_index[E + e];
    if ((unsigned)src >= (unsigned)N) src = 0;
    if ((unsigned)dst >= (unsigned)N) dst = 0;
    int   g  = batch[src] & (BGR - 1);
    float hs = nh[src * TDIR + t];
    float hd = nh[dst * TDIR + t];
    float c  = KSCALE * fmaxf(hs, hd);
    float val = __builtin_amdgcn_rcpf(1.0f + __builtin_amdgcn_exp2f(c - dl));
    myacc[g * SBITS + lane] += val;               // wave-private: plain LDS rmw
  }

#pragma unroll 4
  for (int g = 0; g < BGR; ++g) {
    float vf = myacc[g * SBITS + lane];
    if (vf != 0.0f)
      atomicAdd(&out[(g * SBITS + lane) * TDIR + t], -vf);
  }
}

// ---------------------------------------------------------------------------
// Launch.  Inputs: x[N,3] f32, v[3,32] f32, lin[32] f32, edge_index[2,E] int,
// batch[N] int (sorted), num_graphs (scalar, unused).  d_out: [64,32,32] f32.
// d_ws: nh = N*32 f32 (6.4 MB).
// ---------------------------------------------------------------------------
extern "C" void kernel_launch(void* const* d_in, const int* in_sizes, int n_in,
                              void* d_out, int out_size, void* d_ws,
                              size_t ws_size, hipStream_t stream) {
  const float* x          = (const float*)d_in[0];
  const float* v          = (const float*)d_in[1];
  const float* lin        = (const float*)d_in[2];
  const int*   edge_index = (const int*)d_in[3];
  const int*   batch      = (const int*)d_in[4];

  int N = in_sizes[0] / 3;
  int E = in_sizes[3] / 2;

  float* out = (float*)d_out;
  float* nh  = (float*)d_ws;   // N * 32 floats

  int zb = (out_size + 255) / 256;
  ect_zero<<<zb, 256, 0, stream>>>(out, out_size);

  int waves = (N + 15) / 16;
  int hb = (waves + 7) / 8;                       // 8 waves (256 thr) / block
  ect_heights_wmma<<<hb, 256, 0, stream>>>(x, v, nh, N);

  ect_nodes<<<64, 256, 0, stream>>>(nh, batch, lin, out, N);

  ect_edges<<<EDGE_CHUNKS * TDIR / 8, 256, 0, stream>>>(nh, edge_index, batch,
                                                        lin, out, N, E);
}